// SoftCLT_29669634080790
// MI455X (gfx1250) — compile-verified
//
#include <hip/hip_runtime.h>

#define BB 16
#define T0 256
#define CC 128
#define FF 16

#define BIGV 1.0e10f
#define NEGV -1.0e9f

typedef float v2f __attribute__((ext_vector_type(2)));
typedef float v8f __attribute__((ext_vector_type(8)));

__device__ __forceinline__ v8f wmma_f32_k4(v2f a, v2f b, v8f c) {
  // D = A(16x4, f32) * B(4x16, f32) + C(16x16, f32)
  return __builtin_amdgcn_wmma_f32_16x16x4_f32(false, a, false, b, (short)0, c, false, false);
}

// ---------------- async global->LDS staging (CDNA5 ASYNCcnt path) ----------------
#if defined(__has_builtin)
#if __has_builtin(__builtin_amdgcn_global_load_async_to_lds_b128)
#define HAVE_ASYNC_LDS 1
#endif
#endif
#ifndef HAVE_ASYNC_LDS
#define HAVE_ASYNC_LDS 0
#endif

typedef int v4i_vs __attribute__((vector_size(4 * sizeof(int))));
typedef __attribute__((address_space(1))) v4i_vs* g_v4i_ptr;
typedef __attribute__((address_space(3))) v4i_vs* l_v4i_ptr;

__device__ __forceinline__ void async_copy_b128(void* lds_dst, const void* gsrc) {
#if HAVE_ASYNC_LDS
  // generic LDS pointer: low 32 bits are the LDS byte offset (ISA 10.2)
  __builtin_amdgcn_global_load_async_to_lds_b128(
      (g_v4i_ptr)(unsigned long long)gsrc,
      (l_v4i_ptr)(unsigned)(unsigned long long)lds_dst,
      0, 0);
#else
  *(float4*)lds_dst = *(const float4*)gsrc;
#endif
}

__device__ __forceinline__ void async_wait() {
#if HAVE_ASYNC_LDS
#if __has_builtin(__builtin_amdgcn_s_wait_asynccnt)
  __builtin_amdgcn_s_wait_asynccnt(0);
#else
  asm volatile("s_wait_asynccnt 0" ::: "memory");
#endif
#endif
}

// ---------------- workspace layout (in floats/ints) ----------------
constexpr int OFF_DIST  = 0;                     // 256 f
constexpr int OFF_SL    = 256;                   // 496 f
constexpr int OFF_SR    = 752;                   // 496 f
constexpr int OFF_LEN   = 1248;                  // 16 i
constexpr int OFF_MA    = 1280;                  // 4096 i
constexpr int OFF_MB    = 5376;                  // 4096 i
constexpr int OFF_INSTP = 9472;                  // 1022 f  (511 blocks * 2)
constexpr int OFF_TEMPP = 10496;                 // 2080 f  (1040 blocks * 2)
constexpr int OFF_Z1A   = 12800;                 // 16*128*128 f
constexpr int OFF_Z1B   = OFF_Z1A + 262144;
constexpr int OFF_Z2A   = OFF_Z1B + 262144;
constexpr int OFF_Z2B   = OFF_Z2A + 262144;

// ---------------- lengths + int mask level 0 ----------------
__global__ __launch_bounds__(256) void prep_kernel(const unsigned char* __restrict__ mask,
                                                   int* __restrict__ len,
                                                   int* __restrict__ mlev) {
  __shared__ int sm[256];
  int b = blockIdx.x, t = threadIdx.x;
  int m = mask[b * T0 + t] ? 1 : 0;
  mlev[b * T0 + t] = m;
  sm[t] = m;
  __syncthreads();
  if (t == 0) {
    int s = 0;
    for (int q = 0; q < T0; ++q) s += sm[q];
    len[b] = s;
  }
}

// ---------------- soft-DTW: one workgroup per (i,j) pair ----------------
__global__ __launch_bounds__(256) void dtw_kernel(const float* __restrict__ x,
                                                  const int* __restrict__ len,
                                                  float* __restrict__ dist) {
  __shared__ float xj[T0][FF + 1];
  __shared__ float sqj[T0];
  __shared__ float rbuf[3][T0 + 2];
  int p = blockIdx.x;
  int bi = p >> 4, bj = p & 15;
  int a = threadIdx.x;

  // thread a owns row a of x_i in registers
  float xr[FF];
  float si = 0.f;
  const float* Xi = x + ((size_t)bi * T0 + a) * FF;
  const float* Xj = x + ((size_t)bj * T0 + a) * FF;
#pragma unroll
  for (int k = 0; k < FF; ++k) {
    float v = Xi[k]; xr[k] = v; si += v * v;
  }
  float sj = 0.f;
#pragma unroll
  for (int k = 0; k < FF; ++k) {
    float w = Xj[k]; xj[a][k] = w; sj += w * w;
  }
  sqj[a] = sj;

  rbuf[1][a] = BIGV; rbuf[2][a] = BIGV;
  if (a == 0) {
    rbuf[1][T0] = BIGV; rbuf[2][T0] = BIGV; rbuf[0][T0] = BIGV;
    rbuf[1][0] = 0.0f;   // R(-1,-1) seed
  }
  __syncthreads();

  int li = len[bi] - 1, lj = len[bj] - 1;

  for (int d = 0; d < 2 * T0 - 1; ++d) {
    float* rn = rbuf[d % 3];
    const float* r1 = rbuf[(d + 2) % 3];
    const float* r2 = rbuf[(d + 1) % 3];
    int jj = d - a;
    float val = BIGV;
    if (jj >= 0 && jj < T0) {
      float dot = 0.f;
#pragma unroll
      for (int k = 0; k < FF; ++k) dot += xr[k] * xj[jj][k];
      float cost = si + sqj[jj] - 2.0f * dot;
      float ra = r2[a], rb = r1[a], rc = r1[a + 1];
      float mn = fminf(ra, fminf(rb, rc));
      float s = __expf((mn - ra) * 10.0f) + __expf((mn - rb) * 10.0f) + __expf((mn - rc) * 10.0f);
      float sm = mn - 0.1f * __logf(s);
      val = cost + sm;
      if (a == li && jj == lj) dist[p] = val;   // diags[li+lj, i, j, li+1]
    }
    rn[a + 1] = val;
    if (a == 0) rn[0] = BIGV;
    __syncthreads();
  }
}

// ---------------- soft labels + dup matrices sL/sR (16 x 31) ----------------
__device__ __forceinline__ float slab(const float* dist, int i, int j) {
  float v = dist[i * BB + j] * (1.0f / 50.0f);
  return 1.0f / (1.0f + __expf(-v));   // 2*ALPHA = 1.0
}

__global__ __launch_bounds__(512) void labels_kernel(const float* __restrict__ dist,
                                                     float* __restrict__ sL,
                                                     float* __restrict__ sR) {
  int t = threadIdx.x;
  if (t >= BB * 31) return;
  int i = t / 31, k = t % 31;
  float l, r;
  if (k < 15) { int j2 = (k < i) ? k : k + 1; l = slab(dist, i, j2); }
  else        { l = slab(dist, i, k - 15); }
  if (k < 16) { r = slab(dist, i, k); }
  else        { int j2 = k - 16; j2 = (j2 < i) ? j2 : j2 + 1; r = slab(dist, i, j2); }
  sL[i * 31 + k] = l;
  sR[i * 31 + k] = r;
}

// ---------------- instance contrastive loss (per level) ----------------
// grid = T blocks, 128 threads (4 waves). Each wave computes one 16x16 tile of
// the 32x32 Gram matrix of z_t via V_WMMA_F32_16X16X4_F32.
__global__ __launch_bounds__(128) void inst_kernel(const float* __restrict__ z1,
                                                   const float* __restrict__ z2,
                                                   const int* __restrict__ m,
                                                   const float* __restrict__ sL,
                                                   const float* __restrict__ sR,
                                                   int T, float* __restrict__ part) {
  __shared__ float zz[32][132];
  __shared__ float sim[32][33];
  __shared__ float redL[32], redC[32];
  int t = blockIdx.x;
  int tid = threadIdx.x;

  // async stage 32 rows x 128 floats (1024 x b128 chunks)
  for (int idx = tid; idx < 32 * (CC / 4); idx += 128) {
    int i = idx >> 5;
    int c4 = (idx & 31) * 4;
    const float* src = (i < 16) ? (z1 + ((size_t)i * T + t) * CC + c4)
                                : (z2 + ((size_t)(i - 16) * T + t) * CC + c4);
    async_copy_b128(&zz[i][c4], src);
  }
  async_wait();
  __syncthreads();

  int w = tid >> 5, lane = tid & 31;
  int hi = lane >> 4, lo = lane & 15;
  int wi = (w >> 1) * 16, wj = (w & 1) * 16;

  v8f acc;
#pragma unroll
  for (int q = 0; q < 8; ++q) acc[q] = 0.0f;

  for (int k0 = 0; k0 < CC; k0 += 4) {
    v2f a, b;
    a[0] = zz[wi + lo][k0 + 2 * hi];
    a[1] = zz[wi + lo][k0 + 2 * hi + 1];
    b[0] = zz[wj + lo][k0 + 2 * hi];
    b[1] = zz[wj + lo][k0 + 2 * hi + 1];
    acc = wmma_f32_k4(a, b, acc);
  }
#pragma unroll
  for (int q = 0; q < 8; ++q) sim[wi + q + hi * 8][wj + lo] = acc[q];
  __syncthreads();

  if (tid < 32) {
    int i = tid;
    int vi = m[(i & 15) * T + t];
    float mx = NEGV;
    for (int j = 0; j < 32; ++j) {
      if (j == i) continue;
      int vj = m[(j & 15) * T + t];
      float lg = (vi && vj) ? sim[i][j] : NEGV;
      mx = fmaxf(mx, lg);
    }
    float se = 0.f;
    for (int j = 0; j < 32; ++j) {
      if (j == i) continue;
      int vj = m[(j & 15) * T + t];
      float lg = (vi && vj) ? sim[i][j] : NEGV;
      se += __expf(lg - mx);
    }
    float lse = mx + __logf(se);
    float ls = 0.f, cn = 0.f;
    for (int j = 0; j < 32; ++j) {
      if (j == i) continue;
      int vj = m[(j & 15) * T + t];
      if (vi && vj) {
        float nls = lse - sim[i][j];
        int k = (j < i) ? j : j - 1;
        float wgt = (i < 16) ? sL[i * 31 + k] : sR[(i - 16) * 31 + k];
        ls += nls * wgt;
        cn += 1.f;
      }
    }
    redL[tid] = ls; redC[tid] = cn;
  }
  __syncthreads();
  if (tid == 0) {
    float S = 0.f, Cn = 0.f;
    for (int q = 0; q < 32; ++q) { S += redL[q]; Cn += redC[q]; }
    part[2 * t] = S; part[2 * t + 1] = Cn;
  }
}

// ---------------- time-lag sigmoid weights ----------------
__device__ __forceinline__ float tlv(int i, int j, float sigma) {
  float d = fabsf((float)(i - j)) * sigma;
  float mm = 2.0f / (1.0f + __expf(d));
  return (mm < 1e-6f) ? 0.0f : mm;
}
__device__ __forceinline__ float tLw(int i, int k, int T, float sigma) {
  if (k < T - 1) { int j2 = (k < i) ? k : k + 1; return tlv(i, j2, sigma); }
  return tlv(i, k - (T - 1), sigma);
}
__device__ __forceinline__ float tRw(int i, int k, int T, float sigma) {
  if (k < T) return tlv(i, k, sigma);
  int j2 = k - T; j2 = (j2 < i) ? j2 : j2 + 1;
  return tlv(i, j2, sigma);
}

// ---------------- temporal contrastive loss (2T >= 16) ----------------
// grid = B * (2T/16) blocks, 64 threads (2 waves). Each block computes a
// 16-row band of the (2T x 2T) Gram via WMMA, keeps it in LDS, then does
// masked log-softmax rows with all 64 threads (4 column-groups per row).
__global__ __launch_bounds__(64) void temp_kernel(const float* __restrict__ z1,
                                                  const float* __restrict__ z2,
                                                  const int* __restrict__ m,
                                                  int T, float sigma,
                                                  float* __restrict__ part) {
  __shared__ float ar[16][132];
  __shared__ float bt[2][16][132];
  __shared__ float sim[16][512];
  __shared__ float p4a[4][16], p4b[4][16];
  __shared__ float redL[16], redC[16];
  int TT = 2 * T;
  int tilesPerRow = TT / 16;
  int b = blockIdx.x / tilesPerRow;
  int rt = blockIdx.x % tilesPerRow;
  int R0 = rt * 16;
  int tid = threadIdx.x, w = tid >> 5, lane = tid & 31;
  int hi = lane >> 4, lo = lane & 15;

  // async stage 16 A-rows x 128 floats
  for (int idx = tid; idx < 16 * (CC / 4); idx += 64) {
    int r = idx >> 5;
    int c4 = (idx & 31) * 4;
    int ig = R0 + r;
    const float* src = (ig < T) ? (z1 + ((size_t)b * T + ig) * CC + c4)
                                : (z2 + ((size_t)b * T + (ig - T)) * CC + c4);
    async_copy_b128(&ar[r][c4], src);
  }
  async_wait();
  __syncthreads();

  for (int ct = w; ct < tilesPerRow; ct += 2) {
    // async stage this wave's 16 B-rows
    for (int idx = lane; idx < 16 * (CC / 4); idx += 32) {
      int r = idx >> 5;
      int c4 = (idx & 31) * 4;
      int jg = ct * 16 + r;
      const float* src = (jg < T) ? (z1 + ((size_t)b * T + jg) * CC + c4)
                                  : (z2 + ((size_t)b * T + (jg - T)) * CC + c4);
      async_copy_b128(&bt[w][r][c4], src);
    }
    async_wait();

    v8f acc;
#pragma unroll
    for (int q = 0; q < 8; ++q) acc[q] = 0.0f;
    for (int k0 = 0; k0 < CC; k0 += 4) {
      v2f a, bv;
      a[0]  = ar[lo][k0 + 2 * hi];
      a[1]  = ar[lo][k0 + 2 * hi + 1];
      bv[0] = bt[w][lo][k0 + 2 * hi];
      bv[1] = bt[w][lo][k0 + 2 * hi + 1];
      acc = wmma_f32_k4(a, bv, acc);
    }
#pragma unroll
    for (int q = 0; q < 8; ++q) sim[q + hi * 8][ct * 16 + lo] = acc[q];
  }
  __syncthreads();

  // phase 2: 64 threads = 16 rows x 4 column groups
  int r = tid & 15, qg = tid >> 4;
  int chunk = TT / 4;
  int j0 = qg * chunk, j1 = j0 + chunk;
  int ig = R0 + r;
  int vi = m[b * T + (ig < T ? ig : ig - T)];

  float mx = NEGV;
  for (int j = j0; j < j1; ++j) {
    if (j == ig) continue;
    int vj = m[b * T + (j < T ? j : j - T)];
    float lg = (vi && vj) ? sim[r][j] : NEGV;
    mx = fmaxf(mx, lg);
  }
  p4a[qg][r] = mx;
  __syncthreads();
  float rmx = fmaxf(fmaxf(p4a[0][r], p4a[1][r]), fmaxf(p4a[2][r], p4a[3][r]));

  float se = 0.f;
  for (int j = j0; j < j1; ++j) {
    if (j == ig) continue;
    int vj = m[b * T + (j < T ? j : j - T)];
    float lg = (vi && vj) ? sim[r][j] : NEGV;
    se += __expf(lg - rmx);
  }
  p4b[qg][r] = se;
  __syncthreads();
  float rse = p4b[0][r] + p4b[1][r] + p4b[2][r] + p4b[3][r];
  float lse = rmx + __logf(rse);

  float ls = 0.f, cn = 0.f;
  for (int j = j0; j < j1; ++j) {
    if (j == ig) continue;
    int vj = m[b * T + (j < T ? j : j - T)];
    if (vi && vj) {
      float nls = lse - sim[r][j];
      int k = (j < ig) ? j : j - 1;
      float wgt = (ig < T) ? tLw(ig, k, T, sigma) : tRw(ig - T, k, T, sigma);
      ls += nls * wgt;
      cn += 1.f;
    }
  }
  __syncthreads();           // p4a/p4b reads above are done
  p4a[qg][r] = ls;
  p4b[qg][r] = cn;
  __syncthreads();
  if (tid < 16) {
    redL[tid] = p4a[0][tid] + p4a[1][tid] + p4a[2][tid] + p4a[3][tid];
    redC[tid] = p4b[0][tid] + p4b[1][tid] + p4b[2][tid] + p4b[3][tid];
  }
  __syncthreads();
  if (tid == 0) {
    float S = 0.f, Cn = 0.f;
    for (int q = 0; q < 16; ++q) { S += redL[q]; Cn += redC[q]; }
    part[2 * blockIdx.x] = S; part[2 * blockIdx.x + 1] = Cn;
  }
}

// ---------------- temporal loss for tiny levels (2T = 8 or 4) ----------------
__global__ __launch_bounds__(64) void temp_small_kernel(const float* __restrict__ z1,
                                                        const float* __restrict__ z2,
                                                        const int* __restrict__ m,
                                                        int T, float sigma,
                                                        float* __restrict__ part) {
  __shared__ float zz[8][132];
  __shared__ float sim[8][9];
  __shared__ float redL[8], redC[8];
  int b = blockIdx.x;
  int tid = threadIdx.x;
  int TT = 2 * T;

  for (int idx = tid; idx < TT * CC; idx += 64) {
    int r = idx >> 7, cc = idx & 127;
    const float* src = (r < T) ? (z1 + ((size_t)b * T + r) * CC + cc)
                               : (z2 + ((size_t)b * T + (r - T)) * CC + cc);
    zz[r][cc] = *src;
  }
  __syncthreads();
  if (tid < TT * TT) {
    int i = tid / TT, j = tid % TT;
    float dot = 0.f;
    for (int k = 0; k < CC; ++k) dot += zz[i][k] * zz[j][k];
    sim[i][j] = dot;
  }
  __syncthreads();
  if (tid < TT) {
    int ig = tid;
    int vi = m[b * T + (ig < T ? ig : ig - T)];
    float mx = NEGV;
    for (int j = 0; j < TT; ++j) {
      if (j == ig) continue;
      int vj = m[b * T + (j < T ? j : j - T)];
      float lg = (vi && vj) ? sim[ig][j] : NEGV;
      mx = fmaxf(mx, lg);
    }
    float se = 0.f;
    for (int j = 0; j < TT; ++j) {
      if (j == ig) continue;
      int vj = m[b * T + (j < T ? j : j - T)];
      float lg = (vi && vj) ? sim[ig][j] : NEGV;
      se += __expf(lg - mx);
    }
    float lse = mx + __logf(se);
    float ls = 0.f, cn = 0.f;
    for (int j = 0; j < TT; ++j) {
      if (j == ig) continue;
      int vj = m[b * T + (j < T ? j : j - T)];
      if (vi && vj) {
        float nls = lse - sim[ig][j];
        int k = (j < ig) ? j : j - 1;
        float wgt = (ig < T) ? tLw(ig, k, T, sigma) : tRw(ig - T, k, T, sigma);
        ls += nls * wgt;
        cn += 1.f;
      }
    }
    redL[tid] = ls; redC[tid] = cn;
  }
  __syncthreads();
  if (tid == 0) {
    float S = 0.f, Cn = 0.f;
    for (int q = 0; q < TT; ++q) { S += redL[q]; Cn += redC[q]; }
    part[2 * b] = S; part[2 * b + 1] = Cn;
  }
}

// ---------------- masked max pool (halve T) ----------------
__global__ __launch_bounds__(256) void pool_kernel(const float* __restrict__ zin,
                                                   const int* __restrict__ min_,
                                                   float* __restrict__ zout,
                                                   int* __restrict__ mout,
                                                   int Tin, int writeMask) {
  int T2 = Tin >> 1;
  int idx = blockIdx.x * blockDim.x + threadIdx.x;
  int total = BB * T2 * CC;
  if (idx >= total) return;
  int c = idx & 127;
  int tmp = idx >> 7;
  int tp = tmp % T2;
  int b = tmp / T2;
  int m0 = min_[b * Tin + 2 * tp];
  int m1 = min_[b * Tin + 2 * tp + 1];
  float v0 = m0 ? zin[((size_t)b * Tin + 2 * tp) * CC + c] : -INFINITY;
  float v1 = m1 ? zin[((size_t)b * Tin + 2 * tp + 1) * CC + c] : -INFINITY;
  float mx = fmaxf(v0, v1);
  zout[((size_t)b * T2 + tp) * CC + c] = (m0 || m1) ? mx : 0.0f;
  if (writeMask && c == 0) mout[b * T2 + tp] = (m0 || m1) ? 1 : 0;
}

// ---------------- final deterministic reduction ----------------
__global__ void final_kernel(const float* __restrict__ instP,
                             const float* __restrict__ tempP,
                             float* __restrict__ out) {
  if (threadIdx.x != 0 || blockIdx.x != 0) return;
  float loss = 0.f;
  int off = 0;
  for (int d = 0; d <= 8; ++d) {
    int nb = 256 >> d;
    float S = 0.f, Cn = 0.f;
    for (int q = 0; q < nb; ++q) { S += instP[2 * (off + q)]; Cn += instP[2 * (off + q) + 1]; }
    off += nb;
    loss += 0.5f * (S / fmaxf(Cn, 1.0f));
  }
  off = 0;
  for (int d = 0; d < 8; ++d) {
    int nb = (d <= 5) ? 2 * (256 >> d) : 16;
    float S = 0.f, Cn = 0.f;
    for (int q = 0; q < nb; ++q) { S += tempP[2 * (off + q)]; Cn += tempP[2 * (off + q) + 1]; }
    off += nb;
    loss += 0.5f * (S / fmaxf(Cn, 1.0f));
  }
  out[0] = loss / 9.0f;
}

// ---------------- host launcher ----------------
extern "C" void kernel_launch(void* const* d_in, const int* in_sizes, int n_in,
                              void* d_out, int out_size, void* d_ws, size_t ws_size,
                              hipStream_t stream) {
  const float* z1in = (const float*)d_in[0];
  const float* z2in = (const float*)d_in[1];
  const unsigned char* maskin = (const unsigned char*)d_in[2];
  const float* xin = (const float*)d_in[3];
  float* out = (float*)d_out;
  float* wsF = (float*)d_ws;
  int* wsI = (int*)d_ws;

  float* dist = wsF + OFF_DIST;
  float* sL = wsF + OFF_SL;
  float* sR = wsF + OFF_SR;
  int* len = wsI + OFF_LEN;
  int* mA = wsI + OFF_MA;
  int* mB = wsI + OFF_MB;
  float* instP = wsF + OFF_INSTP;
  float* tempP = wsF + OFF_TEMPP;
  float* z1A = wsF + OFF_Z1A;
  float* z1B = wsF + OFF_Z1B;
  float* z2A = wsF + OFF_Z2A;
  float* z2B = wsF + OFF_Z2B;

  prep_kernel<<<BB, 256, 0, stream>>>(maskin, len, mA);
  dtw_kernel<<<BB * BB, 256, 0, stream>>>(xin, len, dist);
  labels_kernel<<<1, 512, 0, stream>>>(dist, sL, sR);

  const float* z1p = z1in;
  const float* z2p = z2in;
  const int* mp = mA;
  int instOff = 0, tempOff = 0;

  for (int d = 0; d < 8; ++d) {
    int T = 256 >> d;
    float sigma = 2.0f * (float)(1 << d);

    inst_kernel<<<T, 128, 0, stream>>>(z1p, z2p, mp, sL, sR, T, instP + 2 * instOff);
    instOff += T;

    int TT = 2 * T;
    if (TT >= 16) {
      int nb = BB * (TT / 16);
      temp_kernel<<<nb, 64, 0, stream>>>(z1p, z2p, mp, T, sigma, tempP + 2 * tempOff);
      tempOff += nb;
    } else {
      temp_small_kernel<<<BB, 64, 0, stream>>>(z1p, z2p, mp, T, sigma, tempP + 2 * tempOff);
      tempOff += BB;
    }

    float* z1o = (d % 2 == 0) ? z1A : z1B;
    float* z2o = (d % 2 == 0) ? z2A : z2B;
    int* mo = (d % 2 == 0) ? mB : mA;
    int T2 = T >> 1;
    int total = BB * T2 * CC;
    int nb = (total + 255) / 256;
    pool_kernel<<<nb, 256, 0, stream>>>(z1p, mp, z1o, mo, T, 1);
    pool_kernel<<<nb, 256, 0, stream>>>(z2p, mp, z2o, mo, T, 0);
    z1p = z1o; z2p = z2o; mp = mo;
  }

  // final level: T == 1, instance loss only
  inst_kernel<<<1, 128, 0, stream>>>(z1p, z2p, mp, sL, sR, 1, instP + 2 * instOff);
  final_kernel<<<1, 32, 0, stream>>>(instP, tempP, out);
}